// NGCF_37048387895610
// MI455X (gfx1250) — compile-verified
//
#include <hip/hip_runtime.h>
#include <hip/hip_bf16.h>

typedef __attribute__((ext_vector_type(2))) float v2f;
typedef __attribute__((ext_vector_type(8))) float v8f;

#define D 64

// ---------------------------------------------------------------------------
// Utility: zero a float buffer
// ---------------------------------------------------------------------------
__global__ void ngcf_zero_kernel(float* __restrict__ p, long n) {
    long i = (long)blockIdx.x * blockDim.x + threadIdx.x;
    long stride = (long)gridDim.x * blockDim.x;
    for (; i < n; i += stride) p[i] = 0.0f;
}

// ---------------------------------------------------------------------------
// Build emb[0:n_users] = user_emb, emb[n_users:] = item_emb  (float4 granular)
// ---------------------------------------------------------------------------
__global__ void ngcf_init_emb_kernel(const float* __restrict__ ue,
                                     const float* __restrict__ ie,
                                     float* __restrict__ emb,
                                     int n_users, int n_nodes) {
    long t = (long)blockIdx.x * blockDim.x + threadIdx.x;
    int n = (int)(t >> 4);
    int c = (int)(t & 15) << 2;
    if (n >= n_nodes) return;
    float4 v = (n < n_users)
        ? *(const float4*)(ue + (size_t)n * D + c)
        : *(const float4*)(ie + (size_t)(n - n_users) * D + c);
    *(float4*)(emb + (size_t)n * D + c) = v;
}

// ---------------------------------------------------------------------------
// deg[row[e]] += 1  (degree over source nodes, matches reference)
// ---------------------------------------------------------------------------
__global__ void ngcf_deg_kernel(const int* __restrict__ row,
                                float* __restrict__ deg, int E) {
    int e = blockIdx.x * blockDim.x + threadIdx.x;
    if (e < E) atomicAdd(&deg[row[e]], 1.0f);
}

// dinv[n] = deg>0 ? deg^-0.5 : 0   (in place)
__global__ void ngcf_dinv_kernel(float* __restrict__ dinv, int n_nodes) {
    int n = blockIdx.x * blockDim.x + threadIdx.x;
    if (n >= n_nodes) return;
    float d = dinv[n];
    dinv[n] = (d > 0.0f) ? rsqrtf(d) : 0.0f;
}

// ---------------------------------------------------------------------------
// Edge aggregation: agg[col[e]] += dinv[row]*dinv[col] * emb[row[e]]
// 16 lanes per edge, float4 per lane -> coalesced 256B row gather; scatter via
// f32 global atomics that stay resident in the 192MB L2 (agg is 38.4MB).
// ---------------------------------------------------------------------------
__global__ void ngcf_agg_kernel(const int* __restrict__ row,
                                const int* __restrict__ col,
                                const float* __restrict__ dinv,
                                const float* __restrict__ emb,
                                float* __restrict__ agg, int E) {
    long t = (long)blockIdx.x * blockDim.x + threadIdx.x;
    int e = (int)(t >> 4);
    if (e >= E) return;
    int j = (int)(t & 15) << 2;
    int r = row[e];
    int c = col[e];
    float nrm = dinv[r] * dinv[c];
    float4 v = *(const float4*)(emb + (size_t)r * D + j);
    float* dst = agg + (size_t)c * D + j;
    atomicAdd(dst + 0, nrm * v.x);
    atomicAdd(dst + 1, nrm * v.y);
    atomicAdd(dst + 2, nrm * v.z);
    atomicAdd(dst + 3, nrm * v.w);
}

// ---------------------------------------------------------------------------
// Bi-interaction GEMM via V_WMMA_F32_16X16X4_F32 (full fp32 precision):
//   out = leaky_relu( (emb+agg) @ W1^T + b1 + (emb*agg) @ W2^T + b2 )
// Block = 128 threads (4 waves). Block computes 64 rows x 64 cols.
// Each wave owns a 16-row stripe, K=64 consumed in 16 steps of 4.
//
// f32 WMMA fragment layouts (ISA 7.12.2):
//   A 16x4 : lane<16 holds M=lane, K={k0,k0+1}; lane>=16 holds M=lane-16,
//            K={k0+2,k0+3}  (2 VGPRs = v2f)
//   B 4x16 : lane half selects K pair like A; element n = lane&15
//   C 16x16: VGPR r, lane<16 -> M=r, N=lane; lane>=16 -> M=r+8, N=lane-16
// ---------------------------------------------------------------------------
__global__ __launch_bounds__(128) void ngcf_gemm_kernel(
    const float* __restrict__ emb, const float* __restrict__ agg,
    const float* __restrict__ W1, const float* __restrict__ b1,
    const float* __restrict__ W2, const float* __restrict__ b2,
    float* __restrict__ out, int n_nodes) {
    __shared__ float sW1[D * D];   // 16 KB
    __shared__ float sW2[D * D];   // 16 KB
    __shared__ float sX[64 * D];   // 16 KB   emb tile (64 rows)
    __shared__ float sG[64 * D];   // 16 KB   agg tile

    const int tid  = threadIdx.x;
    const int wave = tid >> 5;
    const int lane = tid & 31;
    const int half = lane >> 4;    // K-pair selector
    const int l16  = lane & 15;

    // Stage both 64x64 weight matrices into LDS (float4, coalesced).
    for (int i = tid; i < D * D / 4; i += 128) {
        ((float4*)sW1)[i] = ((const float4*)W1)[i];
        ((float4*)sW2)[i] = ((const float4*)W2)[i];
    }

    // Stage the 64-row emb/agg tile (zero-pad past n_nodes so the WMMA region
    // runs with full EXEC).
    const int row0 = blockIdx.x * 64;
    for (int i = tid; i < 64 * D / 4; i += 128) {
        int r  = i >> 4;
        int cc = (i & 15) << 2;
        int gr = row0 + r;
        float4 xv = make_float4(0.f, 0.f, 0.f, 0.f);
        float4 gv = xv;
        if (gr < n_nodes) {
            xv = *(const float4*)(emb + (size_t)gr * D + cc);
            gv = *(const float4*)(agg + (size_t)gr * D + cc);
        }
        *(float4*)(sX + r * D + cc) = xv;
        *(float4*)(sG + r * D + cc) = gv;
    }
    __syncthreads();

    const int m0 = wave * 16;      // local row base of this wave's stripe

    // Accumulators for the 4 column tiles, initialized with b1+b2 broadcast
    // down the M dimension (bias is per output column only).
    v8f acc[4];
    for (int ct = 0; ct < 4; ++ct) {
        int o = ct * 16 + l16;
        float bb = b1[o] + b2[o];
        for (int r = 0; r < 8; ++r) acc[ct][r] = bb;
    }

    for (int kk = 0; kk < 16; ++kk) {
        const int k0 = (kk << 2) + (half << 1);
        const int arow = m0 + l16;
        float x0 = sX[arow * D + k0], x1 = sX[arow * D + k0 + 1];
        float g0 = sG[arow * D + k0], g1 = sG[arow * D + k0 + 1];
        v2f aS = { x0 + g0, x1 + g1 };   // (emb + agg) fragment
        v2f aP = { x0 * g0, x1 * g1 };   // (emb * agg) fragment
        for (int ct = 0; ct < 4; ++ct) {
            int o = ct * 16 + l16;       // output column
            // B = W^T : B[k][o] = W[o*64 + k]
            v2f bW1 = { sW1[o * D + k0], sW1[o * D + k0 + 1] };
            v2f bW2 = { sW2[o * D + k0], sW2[o * D + k0 + 1] };
            acc[ct] = __builtin_amdgcn_wmma_f32_16x16x4_f32(
                false, aS, false, bW1, (short)0, acc[ct], false, false);
            acc[ct] = __builtin_amdgcn_wmma_f32_16x16x4_f32(
                false, aP, false, bW2, (short)0, acc[ct], false, false);
        }
    }

    // Fused leaky_relu (slope 0.01 < 1  =>  max(v, 0.01v)) + store.
    for (int ct = 0; ct < 4; ++ct) {
        for (int r = 0; r < 8; ++r) {
            int grow = row0 + m0 + r + half * 8;
            if (grow < n_nodes) {
                float v = acc[ct][r];
                v = fmaxf(v, 0.01f * v);
                out[(size_t)grow * D + ct * 16 + l16] = v;
            }
        }
    }
}

// ---------------------------------------------------------------------------
// Partial dot accumulation: out[b] += dot(emb[users[b]], emb[n_users+items[b]])
// (sum over concat == sum of per-layer dots, so no concat buffer needed)
// ---------------------------------------------------------------------------
__global__ void ngcf_dot_kernel(const float* __restrict__ emb,
                                const int* __restrict__ users,
                                const int* __restrict__ items,
                                float* __restrict__ out, int B, int n_users) {
    int b = blockIdx.x * blockDim.x + threadIdx.x;
    if (b >= B) return;
    const float4* u = (const float4*)(emb + (size_t)users[b] * D);
    const float4* v = (const float4*)(emb + (size_t)(n_users + items[b]) * D);
    float s = 0.0f;
    for (int i = 0; i < D / 4; ++i) {
        float4 a = u[i], c = v[i];
        s += a.x * c.x + a.y * c.y + a.z * c.z + a.w * c.w;
    }
    out[b] += s;
}

// ---------------------------------------------------------------------------
// Host orchestration
// ---------------------------------------------------------------------------
extern "C" void kernel_launch(void* const* d_in, const int* in_sizes, int n_in,
                              void* d_out, int out_size, void* d_ws, size_t ws_size,
                              hipStream_t stream) {
    const int*   users = (const int*)d_in[0];
    const int*   items = (const int*)d_in[1];
    const int*   edge  = (const int*)d_in[2];
    const float* ue    = (const float*)d_in[3];
    const float* ie    = (const float*)d_in[4];
    const float* W1    = (const float*)d_in[5];
    const float* b1    = (const float*)d_in[6];
    const float* W2    = (const float*)d_in[7];
    const float* b2    = (const float*)d_in[8];

    const int B       = in_sizes[0];
    const int E       = in_sizes[2] / 2;
    const int n_users = in_sizes[3] / D;
    const int n_items = in_sizes[4] / D;
    const int n_nodes = n_users + n_items;
    const int L       = in_sizes[6] / D;

    const int* row = edge;        // edge_index[0]
    const int* col = edge + E;    // edge_index[1]

    // Workspace layout (floats)
    float* ws = (float*)d_ws;
    size_t embElems = (size_t)n_nodes * D;
    float* embA = ws;
    float* embB = embA + embElems;
    float* agg  = embB + embElems;
    float* dinv = agg + embElems;

    float* out = (float*)d_out;

    const int T = 256;
    const int zeroBlocks = 2048;

    // out = 0
    ngcf_zero_kernel<<<(B + T - 1) / T, T, 0, stream>>>(out, (long)B);

    // emb0 = concat(user_emb, item_emb)
    {
        long tt = (long)n_nodes * 16;
        ngcf_init_emb_kernel<<<(unsigned)((tt + T - 1) / T), T, 0, stream>>>(
            ue, ie, embA, n_users, n_nodes);
    }

    // degree -> dinv
    ngcf_zero_kernel<<<zeroBlocks, T, 0, stream>>>(dinv, (long)n_nodes);
    ngcf_deg_kernel<<<(E + T - 1) / T, T, 0, stream>>>(row, dinv, E);
    ngcf_dinv_kernel<<<(n_nodes + T - 1) / T, T, 0, stream>>>(dinv, n_nodes);

    // layer-0 contribution to the dot products
    ngcf_dot_kernel<<<(B + T - 1) / T, T, 0, stream>>>(embA, users, items, out,
                                                       B, n_users);

    float* cur = embA;
    float* nxt = embB;
    for (int l = 0; l < L; ++l) {
        // agg = 0
        ngcf_zero_kernel<<<zeroBlocks, T, 0, stream>>>(agg, (long)embElems);
        // scatter aggregation
        {
            long tt = (long)E * 16;
            ngcf_agg_kernel<<<(unsigned)((tt + T - 1) / T), T, 0, stream>>>(
                row, col, dinv, cur, agg, E);
        }
        // WMMA bi-interaction GEMM + leaky relu
        {
            unsigned blocks = (unsigned)((n_nodes + 63) / 64);
            ngcf_gemm_kernel<<<blocks, 128, 0, stream>>>(
                cur, agg, W1 + (size_t)l * D * D, b1 + (size_t)l * D,
                W2 + (size_t)l * D * D, b2 + (size_t)l * D, nxt, n_nodes);
        }
        // accumulate this layer's dot contribution
        ngcf_dot_kernel<<<(B + T - 1) / T, T, 0, stream>>>(nxt, users, items,
                                                           out, B, n_users);
        float* tmp = cur; cur = nxt; nxt = tmp;
    }
}